// EfronLossPenalty_74380243632284
// MI455X (gfx1250) — compile-verified
//
#include <hip/hip_runtime.h>
#include <math.h>

// ---------------------------------------------------------------------------
// Efron Cox partial-likelihood loss, MI455X (gfx1250, wave32).
//
// Phase 1 (bandwidth bound, ~96MB input -> 4.3us floor @ 23.3 TB/s):
//   per-workgroup LDS histogram over 8192 time bins (S, D, E, L components,
//   128KB LDS), two vec4 chunks in flight per loop trip for MLP, flushed to
//   per-block partials (no global atomics -> deterministic).
// Phase 2: sum partials serially, in block order, into final bins.
// Phase 3: suffix scan of S over 8192 bins. The 256 inter-chunk partial sums
//   are scanned with V_WMMA_F32_16X16X4_F32:  scan(M) = M*U + Lstrict*(M*J)
//   (U = upper-triangular ones, J = all ones), split into K=4 chunks.
// Phase 4: per-bin Efron terms  sum_{l<d} log(max(rss - (l/d)*ers, eps))
//   (d==1 -> log(rss+eps)), block-reduced into blog[t].
// Phase 5: final scalar reduce + divide by max(num_events, 1).
// ---------------------------------------------------------------------------

#define NBINS 8192
#define TPB   256
#define EFRON_EPS 1e-12f

typedef int   v4i  __attribute__((ext_vector_type(4)));
typedef float v4f  __attribute__((ext_vector_type(4)));
typedef float v2f_ __attribute__((ext_vector_type(2)));
typedef float v8f_ __attribute__((ext_vector_type(8)));

// workspace layout (float offsets)
#define OFF_S    0
#define OFF_D    (NBINS)
#define OFF_E    (2 * NBINS)
#define OFF_L    (3 * NBINS)
#define OFF_RSS  (4 * NBINS)
#define OFF_BLOG (5 * NBINS)
#define OFF_SCAL (6 * NBINS)      // [0]=num_events, [1]=sum(e*lr)
#define OFF_PART (7 * NBINS)      // G x (4*NBINS) per-block partials

// ---------------------------------------------------------------------------
// Phase 1: streaming histogram. 4*NBINS floats of dynamic LDS.
// ---------------------------------------------------------------------------
__device__ __forceinline__ void hist_accum(float* ldsS, float* ldsD,
                                           float* ldsE, float* ldsL,
                                           v4i tv, v4i ev, v4f lv)
{
#pragma unroll
  for (int c = 0; c < 4; ++c) {
    const int   t = tv[c] & (NBINS - 1);
    const float l = lv[c];
    const float r = __expf(l);
    atomicAdd(&ldsS[t], r);
    if (ev[c]) {
      atomicAdd(&ldsD[t], 1.0f);
      atomicAdd(&ldsE[t], r);
      atomicAdd(&ldsL[t], l);
    }
  }
}

__global__ __launch_bounds__(TPB) void hist_kernel(
    const int* __restrict__ times, const int* __restrict__ events,
    const float* __restrict__ logr, float* __restrict__ partials, int n)
{
  extern __shared__ float lds[];          // 4*NBINS floats = 128 KB
  const int tid = threadIdx.x;
  for (int i = tid; i < 4 * NBINS; i += TPB) lds[i] = 0.0f;
  __syncthreads();
  float* ldsS = lds;
  float* ldsD = lds + NBINS;
  float* ldsE = lds + 2 * NBINS;
  float* ldsL = lds + 3 * NBINS;

  const int nvec  = n >> 2;
  const int gsize = gridDim.x * TPB;
  const v4i* tp = (const v4i*)times;
  const v4i* ep = (const v4i*)events;
  const v4f* rp = (const v4f*)logr;

  int i = blockIdx.x * TPB + tid;
  // two independent vec4 chunks per trip: all 6 global_load_b128 issued
  // before any LDS work -> more bytes in flight per wave.
  for (; i + gsize < nvec; i += 2 * gsize) {
    const int j = i + gsize;
    v4i tv0 = __builtin_nontemporal_load(tp + i);
    v4i ev0 = __builtin_nontemporal_load(ep + i);
    v4f lv0 = __builtin_nontemporal_load(rp + i);
    v4i tv1 = __builtin_nontemporal_load(tp + j);
    v4i ev1 = __builtin_nontemporal_load(ep + j);
    v4f lv1 = __builtin_nontemporal_load(rp + j);
    hist_accum(ldsS, ldsD, ldsE, ldsL, tv0, ev0, lv0);
    hist_accum(ldsS, ldsD, ldsE, ldsL, tv1, ev1, lv1);
  }
  for (; i < nvec; i += gsize) {
    v4i tv = __builtin_nontemporal_load(tp + i);
    v4i ev = __builtin_nontemporal_load(ep + i);
    v4f lv = __builtin_nontemporal_load(rp + i);
    hist_accum(ldsS, ldsD, ldsE, ldsL, tv, ev, lv);
  }
  // scalar tail (n % 4 != 0) handled by block 0
  if (blockIdx.x == 0) {
    for (int s = (nvec << 2) + tid; s < n; s += TPB) {
      const int   t = times[s] & (NBINS - 1);
      const float l = logr[s];
      const float r = __expf(l);
      atomicAdd(&ldsS[t], r);
      if (events[s]) {
        atomicAdd(&ldsD[t], 1.0f);
        atomicAdd(&ldsE[t], r);
        atomicAdd(&ldsL[t], l);
      }
    }
  }
  __syncthreads();
  float* my = partials + (size_t)blockIdx.x * (4 * NBINS);
  for (int k = tid; k < 4 * NBINS; k += TPB)
    __builtin_nontemporal_store(lds[k], my + k);
}

// ---------------------------------------------------------------------------
// Phase 2: combine per-block partials -> bins (deterministic order).
// ---------------------------------------------------------------------------
__global__ __launch_bounds__(TPB) void reduce_kernel(
    const float* __restrict__ partials, float* __restrict__ bins, int G)
{
  const int i = blockIdx.x * TPB + threadIdx.x;
  if (i >= 4 * NBINS) return;
  float s = 0.0f;
  for (int g = 0; g < G; ++g) s += partials[(size_t)g * (4 * NBINS) + i];
  bins[i] = s;
}

// ---------------------------------------------------------------------------
// Phase 3: suffix scan of S over bins (single workgroup, 32 bins/thread).
// Inter-chunk scan of 256 chunk totals done with V_WMMA_F32_16X16X4_F32.
// Also computes num_events = sum(D), sumL = sum(L).
// ---------------------------------------------------------------------------
__global__ __launch_bounds__(TPB) void scan_kernel(float* __restrict__ ws)
{
  const float* S = ws + OFF_S;
  const float* D = ws + OFF_D;
  const float* L = ws + OFF_L;
  float* rss  = ws + OFF_RSS;
  float* scal = ws + OFF_SCAL;

  __shared__ float totals[256];
  __shared__ float qmat[256];
  __shared__ float incl[256];
  __shared__ float redD[256];
  __shared__ float redL[256];

  const int tid  = threadIdx.x;
  const int CH   = NBINS / TPB;          // 32
  const int base = tid * CH;

  // pass 1: per-thread chunk totals (in descending-time order => suffix scan)
  float loc = 0.0f, locD = 0.0f, locL = 0.0f;
  for (int k = 0; k < CH; ++k) {
    loc  += S[NBINS - 1 - (base + k)];
    locD += D[base + k];
    locL += L[base + k];
  }
  totals[tid] = loc; redD[tid] = locD; redL[tid] = locL;
  __syncthreads();

  for (int s = 128; s > 0; s >>= 1) {
    if (tid < s) { redD[tid] += redD[tid + s]; redL[tid] += redL[tid + s]; }
    __syncthreads();
  }

  // WMMA inclusive scan of totals[256] on wave 0.
  // View totals as 16x16 row-major M; scan(M) = M*U + Lstrict*(M*J).
  v8f_ P = {0, 0, 0, 0, 0, 0, 0, 0};
  const int half = (tid >> 4) & 1;       // K-half / M-half selector
  const int l15  = tid & 15;
  if (tid < 32) {
    v8f_ R = {0, 0, 0, 0, 0, 0, 0, 0};
#pragma unroll
    for (int k = 0; k < 4; ++k) {
      const int k0 = 4 * k + 2 * half;   // K index for reg 0 in this lane
      const int k1 = k0 + 1;             // K index for reg 1
      // A = M (16x4 slice): A[lane][reg] = M[l15][K]
      v2f_ a;  a.x  = totals[l15 * 16 + k0]; a.y = totals[l15 * 16 + k1];
      // B = U (4x16 slice): U[K][l15] = (K <= l15)
      v2f_ bu; bu.x = (k0 <= l15) ? 1.0f : 0.0f;
               bu.y = (k1 <= l15) ? 1.0f : 0.0f;
      v2f_ bj; bj.x = 1.0f; bj.y = 1.0f; // B = J (all ones)
      P = __builtin_amdgcn_wmma_f32_16x16x4_f32(false, a, false, bu,
                                                (short)0, P, false, false);
      R = __builtin_amdgcn_wmma_f32_16x16x4_f32(false, a, false, bj,
                                                (short)0, R, false, false);
    }
    // stage R = M*J through LDS to re-load it in B layout
#pragma unroll
    for (int j = 0; j < 8; ++j) qmat[(j + 8 * half) * 16 + l15] = R[j];
  }
  __syncthreads();
  if (tid < 32) {
#pragma unroll
    for (int k = 0; k < 4; ++k) {
      const int k0 = 4 * k + 2 * half;
      const int k1 = k0 + 1;
      // A = Lstrict: Lstrict[l15][K] = (K < l15)
      v2f_ al; al.x = (k0 < l15) ? 1.0f : 0.0f;
               al.y = (k1 < l15) ? 1.0f : 0.0f;
      // B = R: R[K][l15]
      v2f_ br; br.x = qmat[k0 * 16 + l15]; br.y = qmat[k1 * 16 + l15];
      P = __builtin_amdgcn_wmma_f32_16x16x4_f32(false, al, false, br,
                                                (short)0, P, false, false);
    }
#pragma unroll
    for (int j = 0; j < 8; ++j) incl[(j + 8 * half) * 16 + l15] = P[j];
  }
  __syncthreads();

  // pass 2: exclusive chunk offset + local running suffix sum
  float run = (tid == 0) ? 0.0f : incl[tid - 1];
  for (int k = 0; k < CH; ++k) {
    const int t = NBINS - 1 - (base + k);
    run += S[t];
    rss[t] = run;
  }
  if (tid == 0) { scal[0] = redD[0]; scal[1] = redL[0]; }
}

// ---------------------------------------------------------------------------
// Phase 4: per-bin Efron log terms. One block per bin.
// ---------------------------------------------------------------------------
__global__ __launch_bounds__(TPB) void loss_kernel(float* __restrict__ ws)
{
  const float* D   = ws + OFF_D;
  const float* E   = ws + OFF_E;
  const float* rss = ws + OFF_RSS;
  float* blog = ws + OFF_BLOG;
  __shared__ float red[TPB];

  const int t   = blockIdx.x;
  const int tid = threadIdx.x;
  const float d  = D[t];
  const int   di = (int)(d + 0.5f);
  const float rv  = rss[t];
  const float ers = E[t];

  float loc = 0.0f;
  if (di == 1) {
    if (tid == 0) loc = __logf(rv + EFRON_EPS);
  } else if (di > 1) {
    const float invd = 1.0f / d;
    for (int l = tid; l < di; l += TPB) {
      float arg = rv - ((float)l * invd) * ers;
      loc += __logf(fmaxf(arg, EFRON_EPS));
    }
  }
  red[tid] = loc;
  __syncthreads();
  for (int s = TPB / 2; s > 0; s >>= 1) {
    if (tid < s) red[tid] += red[tid + s];
    __syncthreads();
  }
  if (tid == 0) blog[t] = red[0];       // always written (0 for d==0 bins)
}

// ---------------------------------------------------------------------------
// Phase 5: final scalar.
// ---------------------------------------------------------------------------
__global__ __launch_bounds__(TPB) void finalize_kernel(
    const float* __restrict__ ws, float* __restrict__ out)
{
  __shared__ float red[TPB];
  const float* blog = ws + OFF_BLOG;
  const float* scal = ws + OFF_SCAL;
  const int tid = threadIdx.x;
  float s = 0.0f;
  for (int i = tid; i < NBINS; i += TPB) s += blog[i];
  red[tid] = s;
  __syncthreads();
  for (int st = TPB / 2; st > 0; st >>= 1) {
    if (tid < st) red[tid] += red[tid + st];
    __syncthreads();
  }
  if (tid == 0) {
    const float num = fmaxf(scal[0], 1.0f);
    out[0] = (red[0] - scal[1]) / num;
  }
}

// ---------------------------------------------------------------------------
extern "C" void kernel_launch(void* const* d_in, const int* in_sizes, int n_in,
                              void* d_out, int out_size, void* d_ws, size_t ws_size,
                              hipStream_t stream)
{
  (void)n_in; (void)out_size;
  const int*   times  = (const int*)d_in[0];
  const int*   events = (const int*)d_in[1];
  const float* logr   = (const float*)d_in[2];
  float* ws  = (float*)d_ws;
  float* out = (float*)d_out;
  const int n = in_sizes[0];

  // workgroup count for the streaming phase, limited by partials capacity
  // (deterministic: depends only on ws_size).
  const size_t avail = ws_size / sizeof(float);
  int G = 1;
  if (avail > (size_t)OFF_PART + 4 * NBINS) {
    size_t g = (avail - OFF_PART) / (4 * NBINS);
    G = (int)(g > 256 ? 256 : g);
    if (G < 1) G = 1;
  }
  float* partials = ws + OFF_PART;

  hist_kernel<<<G, TPB, 4 * NBINS * sizeof(float), stream>>>(
      times, events, logr, partials, n);
  reduce_kernel<<<(4 * NBINS) / TPB, TPB, 0, stream>>>(partials, ws, G);
  scan_kernel<<<1, TPB, 0, stream>>>(ws);
  loss_kernel<<<NBINS, TPB, 0, stream>>>(ws);
  finalize_kernel<<<1, TPB, 0, stream>>>(ws, out);
}